// HAN_67877663146321
// MI455X (gfx1250) — compile-verified
//
#include <hip/hip_runtime.h>
#include <math.h>

// ---------------- problem constants (fixed by the reference) ----------------
constexpr int N_NODES = 100000;
constexpr int E_EDGES = 800000;
constexpr int HID     = 64;
constexpr int HEADS   = 4;
constexpr int RELS    = 3;
constexpr int OUTF    = HEADS * HID;          // 256
constexpr int ETOT    = E_EDGES + N_NODES;    // edges + self loops
constexpr float NEG_SLOPE = 0.2f;
constexpr float NEG_INF_F = -1e9f;
constexpr int MTILES  = N_NODES / 16;         // 6250 (exact)
constexpr int NTILES_SEM = (N_NODES * RELS) / 16; // 18750 (exact)

typedef __attribute__((ext_vector_type(2))) float v2f;
typedef __attribute__((ext_vector_type(8))) float v8f;

// order-preserving float <-> uint encoding for atomicMax on floats
__device__ __forceinline__ unsigned fenc(float f) {
    unsigned u = __float_as_uint(f);
    return (u & 0x80000000u) ? ~u : (u | 0x80000000u);
}
__device__ __forceinline__ float fdec(unsigned e) {
    return (e & 0x80000000u) ? __uint_as_float(e ^ 0x80000000u)
                             : __uint_as_float(~e);
}

__device__ __forceinline__ bool edge_get(const int* __restrict__ ei,
                                         const int* __restrict__ et,
                                         int rel, int idx, int& s, int& d) {
    if (idx < E_EDGES) {
        s = ei[idx];
        d = ei[E_EDGES + idx];
        return et[idx] == rel;
    }
    s = d = idx - E_EDGES;   // self loop, always active
    return true;
}

// ---------------- init kernels ----------------
__global__ void k_init_z(const float* __restrict__ bias, float* __restrict__ z) {
    int i = blockIdx.x * blockDim.x + threadIdx.x;
    if (i >= N_NODES * RELS * HID) return;
    int c = i % HID;
    int r = (i / HID) % RELS;
    z[i] = bias[r * HID + c];
}

__global__ void k_zero(float* __restrict__ wsum) {
    int i = threadIdx.x;
    if (i < 8) wsum[i] = 0.0f;
}

__global__ void k_init_md(unsigned* __restrict__ menc, float* __restrict__ den) {
    int i = blockIdx.x * blockDim.x + threadIdx.x;
    if (i >= N_NODES * HEADS) return;
    menc[i] = fenc(NEG_INF_F);
    den[i]  = 0.0f;
}

// ---------------- GEMM: xw = emb[x] @ W[r]  ([N,64] x [64,256]) ----------------
// wave32 per 16x16 output tile, V_WMMA_F32_16X16X4_F32, K looped 64/4 = 16 steps.
__global__ void k_gemm_xw(const int* __restrict__ x,
                          const float* __restrict__ emb,
                          const float* __restrict__ W,     // [64,256] row-major
                          float* __restrict__ xw) {        // [N,256]
    int mtile = blockIdx.x * blockDim.y + threadIdx.y;     // wave-uniform
    if (mtile >= MTILES) return;
    int ntile = blockIdx.y;                                 // 0..15
    int lane  = threadIdx.x;
    int half  = lane >> 4;
    int l15   = lane & 15;

    int m    = mtile * 16 + l15;
    int node = x[m];
    const float* arow = emb + (size_t)node * HID;
    int cc = ntile * 16 + l15;

    v8f acc = {};
    #pragma unroll
    for (int k0 = 0; k0 < HID; k0 += 4) {
        int kb = k0 + half * 2;
        v2f a, b;
        a.x = arow[kb];
        a.y = arow[kb + 1];
        b.x = W[(size_t)kb * OUTF + cc];
        b.y = W[(size_t)(kb + 1) * OUTF + cc];
        acc = __builtin_amdgcn_wmma_f32_16x16x4_f32(
            false, a, false, b, (short)0, acc, false, false);
    }
    #pragma unroll
    for (int j = 0; j < 8; j++) {
        int mm = mtile * 16 + j + half * 8;
        xw[(size_t)mm * OUTF + cc] = acc[j];
    }
}

// ---------------- per-node attention logits ----------------
__global__ void k_alpha(const float* __restrict__ xw,
                        const float* __restrict__ as,   // [4,64] for this rel
                        const float* __restrict__ ad,
                        float* __restrict__ asrc, float* __restrict__ adst) {
    int t = blockIdx.x * blockDim.x + threadIdx.x;
    if (t >= N_NODES * HEADS) return;
    int n = t >> 2, h = t & 3;
    const float* xr = xw + (size_t)n * OUTF + h * HID;
    const float* a1 = as + h * HID;
    const float* a2 = ad + h * HID;
    float s1 = 0.f, s2 = 0.f;
    #pragma unroll 8
    for (int c = 0; c < HID; c++) {
        float v = xr[c];
        s1 += v * a1[c];
        s2 += v * a2[c];
    }
    asrc[t] = s1;
    adst[t] = s2;
}

// ---------------- segment softmax over destination nodes ----------------
__global__ void k_edge_max(const int* __restrict__ ei, const int* __restrict__ et,
                           int rel, const float* __restrict__ asrc,
                           const float* __restrict__ adst,
                           unsigned* __restrict__ menc) {
    int idx = blockIdx.x * blockDim.x + threadIdx.x;
    if (idx >= ETOT) return;
    int s, d;
    if (!edge_get(ei, et, rel, idx, s, d)) return;
    #pragma unroll
    for (int h = 0; h < HEADS; h++) {
        float e = asrc[s * HEADS + h] + adst[d * HEADS + h];
        e = e > 0.f ? e : NEG_SLOPE * e;
        atomicMax(&menc[d * HEADS + h], fenc(e));
    }
}

__global__ void k_edge_denom(const int* __restrict__ ei, const int* __restrict__ et,
                             int rel, const float* __restrict__ asrc,
                             const float* __restrict__ adst,
                             const unsigned* __restrict__ menc,
                             float* __restrict__ den) {
    int idx = blockIdx.x * blockDim.x + threadIdx.x;
    if (idx >= ETOT) return;
    int s, d;
    if (!edge_get(ei, et, rel, idx, s, d)) return;
    #pragma unroll
    for (int h = 0; h < HEADS; h++) {
        float e = asrc[s * HEADS + h] + adst[d * HEADS + h];
        e = e > 0.f ? e : NEG_SLOPE * e;
        float p = expf(e - fdec(menc[d * HEADS + h]));
        atomicAdd(&den[d * HEADS + h], p);
    }
}

// wave per edge: 32 lanes cover 4 heads x 64 ch (8 ch per lane).
// Adds xw[src]*alpha/HEADS directly into z[:, rel, :] (head-mean fused).
__global__ void k_edge_agg(const int* __restrict__ ei, const int* __restrict__ et,
                           int rel, const float* __restrict__ asrc,
                           const float* __restrict__ adst,
                           const unsigned* __restrict__ menc,
                           const float* __restrict__ den,
                           const float* __restrict__ xw,
                           float* __restrict__ z) {
    int edge = blockIdx.x * blockDim.y + threadIdx.y;   // wave-uniform
    if (edge >= ETOT) return;
    int s, d;
    if (!edge_get(ei, et, rel, edge, s, d)) return;     // uniform per wave
    int lane = threadIdx.x;
    int h  = lane >> 3;
    int c0 = (lane & 7) * 8;
    float e = asrc[s * HEADS + h] + adst[d * HEADS + h];
    e = e > 0.f ? e : NEG_SLOPE * e;
    float p = expf(e - fdec(menc[d * HEADS + h]));
    float alpha = p / den[d * HEADS + h];
    float sc = alpha * (1.0f / HEADS);
    const float* xr = xw + (size_t)s * OUTF + h * HID;
    float* zr = z + ((size_t)d * RELS + rel) * HID;
    #pragma unroll
    for (int j = 0; j < 8; j++)
        atomicAdd(&zr[c0 + j], xr[c0 + j] * sc);
}

// ---------------- semantic attention GEMM: tanh(Z@W1+b1)@W2, summed per rel --
// Z viewed as [N*R, 64]; wave per 16-row tile; 4 n-tiles of WMMA with fused
// bias+tanh+W2-dot epilogue, shuffle-reduced, LDS partials, 3 atomics/block.
__global__ void k_sem(const float* __restrict__ z,
                      const float* __restrict__ W1,   // [64,64]
                      const float* __restrict__ b1,   // [64]
                      const float* __restrict__ W2,   // [64]
                      float* __restrict__ wsum) {     // [3]
    __shared__ float part[RELS];
    int lt = threadIdx.y * 32 + threadIdx.x;
    if (lt < RELS) part[lt] = 0.0f;
    __syncthreads();

    int tile = blockIdx.x * blockDim.y + threadIdx.y;   // wave-uniform
    int lane = threadIdx.x;
    int half = lane >> 4;
    int l15  = lane & 15;

    if (tile < NTILES_SEM) {
        int rowBase = tile * 16;
        const float* arow = z + (size_t)(rowBase + l15) * HID;
        float rowacc[8];
        #pragma unroll
        for (int j = 0; j < 8; j++) rowacc[j] = 0.0f;

        #pragma unroll
        for (int nt = 0; nt < 4; nt++) {
            int cc = nt * 16 + l15;
            v8f acc = {};
            #pragma unroll
            for (int k0 = 0; k0 < HID; k0 += 4) {
                int kb = k0 + half * 2;
                v2f a, b;
                a.x = arow[kb];
                a.y = arow[kb + 1];
                b.x = W1[(size_t)kb * HID + cc];
                b.y = W1[(size_t)(kb + 1) * HID + cc];
                acc = __builtin_amdgcn_wmma_f32_16x16x4_f32(
                    false, a, false, b, (short)0, acc, false, false);
            }
            float b1v = b1[cc];
            float w2v = W2[cc];
            #pragma unroll
            for (int j = 0; j < 8; j++)
                rowacc[j] += tanhf(acc[j] + b1v) * w2v;
        }
        // reduce the 16 column-partials per row (within each 16-lane half)
        #pragma unroll
        for (int j = 0; j < 8; j++) {
            #pragma unroll
            for (int off = 8; off >= 1; off >>= 1)
                rowacc[j] += __shfl_xor(rowacc[j], off, 32);
        }
        if (l15 == 0) {   // lane 0 -> rows base+0..7, lane 16 -> rows base+8..15
            #pragma unroll
            for (int j = 0; j < 8; j++) {
                int row = rowBase + j + half * 8;   // row = n*RELS + r
                atomicAdd(&part[row % RELS], rowacc[j]);
            }
        }
    }
    __syncthreads();
    if (lt < RELS) atomicAdd(&wsum[lt], part[lt]);
}

__global__ void k_beta(const float* __restrict__ wsum, float* __restrict__ beta) {
    if (blockIdx.x == 0 && threadIdx.x == 0) {
        float m0 = wsum[0] / (float)N_NODES;
        float m1 = wsum[1] / (float)N_NODES;
        float m2 = wsum[2] / (float)N_NODES;
        float mx = fmaxf(m0, fmaxf(m1, m2));
        float e0 = expf(m0 - mx), e1 = expf(m1 - mx), e2 = expf(m2 - mx);
        float inv = 1.0f / (e0 + e1 + e2);
        beta[0] = e0 * inv;
        beta[1] = e1 * inv;
        beta[2] = e2 * inv;
    }
}

__global__ void k_final(const float* __restrict__ z, const float* __restrict__ beta,
                        float* __restrict__ out) {
    int i = blockIdx.x * blockDim.x + threadIdx.x;
    if (i >= N_NODES * HID) return;
    int n = i / HID, c = i % HID;
    const float* zr = z + (size_t)n * RELS * HID;
    float v = 0.f;
    #pragma unroll
    for (int r = 0; r < RELS; r++)
        v += beta[r] * zr[r * HID + c];
    out[i] = fmaxf(v, 0.0f);
}

// ---------------- host launch ----------------
extern "C" void kernel_launch(void* const* d_in, const int* in_sizes, int n_in,
                              void* d_out, int out_size, void* d_ws, size_t ws_size,
                              hipStream_t stream) {
    const int*   x       = (const int*)d_in[0];
    const int*   ei      = (const int*)d_in[1];
    const int*   et      = (const int*)d_in[2];
    const float* emb     = (const float*)d_in[3];
    const float* gat_W   = (const float*)d_in[4];   // [3,64,256]
    const float* att_src = (const float*)d_in[5];   // [3,4,64]
    const float* att_dst = (const float*)d_in[6];
    const float* gat_b   = (const float*)d_in[7];   // [3,64]
    const float* sem_W1  = (const float*)d_in[8];   // [64,64]
    const float* sem_b1  = (const float*)d_in[9];   // [64]
    const float* sem_W2  = (const float*)d_in[10];  // [64,1]
    float* out = (float*)d_out;

    float* ws    = (float*)d_ws;
    float* xw    = ws;                                       // N*256
    float* asrc  = xw   + (size_t)N_NODES * OUTF;            // N*4
    float* adst  = asrc + (size_t)N_NODES * HEADS;           // N*4
    unsigned* menc = (unsigned*)(adst + (size_t)N_NODES * HEADS); // N*4
    float* den   = (float*)menc + (size_t)N_NODES * HEADS;   // N*4
    float* z     = den  + (size_t)N_NODES * HEADS;           // N*3*64
    float* wsum  = z    + (size_t)N_NODES * RELS * HID;      // 8
    float* beta  = wsum + 8;                                 // 3

    k_init_z<<<(N_NODES * RELS * HID + 255) / 256, 256, 0, stream>>>(gat_b, z);
    k_zero<<<1, 64, 0, stream>>>(wsum);

    for (int r = 0; r < RELS; r++) {
        k_gemm_xw<<<dim3((MTILES + 7) / 8, OUTF / 16), dim3(32, 8), 0, stream>>>(
            x, emb, gat_W + (size_t)r * HID * OUTF, xw);
        k_alpha<<<(N_NODES * HEADS + 255) / 256, 256, 0, stream>>>(
            xw, att_src + (size_t)r * HEADS * HID, att_dst + (size_t)r * HEADS * HID,
            asrc, adst);
        k_init_md<<<(N_NODES * HEADS + 255) / 256, 256, 0, stream>>>(menc, den);
        k_edge_max<<<(ETOT + 255) / 256, 256, 0, stream>>>(ei, et, r, asrc, adst, menc);
        k_edge_denom<<<(ETOT + 255) / 256, 256, 0, stream>>>(ei, et, r, asrc, adst, menc, den);
        k_edge_agg<<<dim3((ETOT + 7) / 8), dim3(32, 8), 0, stream>>>(
            ei, et, r, asrc, adst, menc, den, xw, z);
    }

    k_sem<<<(NTILES_SEM + 7) / 8, dim3(32, 8), 0, stream>>>(z, sem_W1, sem_b1, sem_W2, wsum);
    k_beta<<<1, 1, 0, stream>>>(wsum, beta);
    k_final<<<(N_NODES * HID + 255) / 256, 256, 0, stream>>>(z, beta, out);
}